// BahdanauAttention_90159953478440
// MI455X (gfx1250) — compile-verified
//
#include <hip/hip_runtime.h>
#include <hip/hip_bf16.h>
#include <math.h>

// Problem constants (from reference): G=3, B=32, T=1024, D=512, U=512
constexpr int G = 3;
constexpr int Bc = 32;
constexpr int T = 1024;
constexpr int D = 512;
constexpr int U = 512;

constexpr int TT  = 64;        // T-tile rows per workgroup in score kernel
constexpr int LDA = D + 4;     // padded LDS row stride (floats) -> conflict-free WMMA A reads

typedef __attribute__((ext_vector_type(2))) float v2f;
typedef __attribute__((ext_vector_type(8))) float v8f;

// CDNA5 async global->LDS copy path (ASYNCcnt-tracked, no VGPR staging).
// Param types per hipcc diagnostic: (AS1 v4i*, AS3 v4i*, imm offset, imm cpol).
typedef int gv4i __attribute__((vector_size(16)));
typedef __attribute__((address_space(1))) gv4i* as1_v4i_ptr;
typedef __attribute__((address_space(3))) gv4i* as3_v4i_ptr;

#if defined(__has_builtin)
#if __has_builtin(__builtin_amdgcn_global_load_async_to_lds_b128)
#define USE_ASYNC_STAGE 1
#endif
#endif

__device__ __forceinline__ void wait_async_zero() {
#if defined(__has_builtin)
#if __has_builtin(__builtin_amdgcn_s_wait_asynccnt)
    __builtin_amdgcn_s_wait_asynccnt(0);
    return;
#else
    asm volatile("s_wait_asynccnt 0x0" ::: "memory");
    return;
#endif
#else
    asm volatile("s_wait_asynccnt 0x0" ::: "memory");
#endif
}

// ---------------------------------------------------------------------------
// Kernel 1: qp[g,b,u] = query[g,b,:] . W1[g,:,u] + b1[g,u]
// M = G*B = 96 rows (each 16-row tile lies inside a single g since B=32),
// N = U = 512, K = D = 512. One wave per 16x16 output tile, WMMA f32 16x16x4.
// ---------------------------------------------------------------------------
__global__ __launch_bounds__(32) void qp_wmma_kernel(
    const float* __restrict__ query,   // [G*B, D]
    const float* __restrict__ W1,      // [G, D, U]
    const float* __restrict__ b1,      // [G, U]
    float* __restrict__ qp)            // [G*B, U]
{
    const int mtile = blockIdx.x;          // 0..5
    const int ntile = blockIdx.y;          // 0..31
    const int g     = mtile >> 1;          // 2 m-tiles per g (B=32)
    const int lane  = threadIdx.x;
    const int r     = lane & 15;
    const int kh    = lane >> 4;           // K-half select (0: K=0,1  1: K=2,3)

    const float* Arow = query + (size_t)(mtile * 16 + r) * D;
    const float* Bg   = W1 + (size_t)g * D * U;
    const int    col  = ntile * 16 + r;

    v8f acc = {};
    #pragma unroll 2
    for (int k = 0; k < D; k += 4) {
        const int ka = k + 2 * kh;
        v2f a; a.x = Arow[ka];                 a.y = Arow[ka + 1];
        v2f b; b.x = Bg[(size_t)ka * U + col]; b.y = Bg[(size_t)(ka + 1) * U + col];
        acc = __builtin_amdgcn_wmma_f32_16x16x4_f32(false, a, false, b,
                                                    (short)0, acc, false, false);
    }

    // D-matrix layout: col = lane&15, rows = (lane>>4)*8 + rr
    const float bias  = b1[(size_t)g * U + col];
    const int   rbase = kh * 8;
    #pragma unroll
    for (int rr = 0; rr < 8; ++rr) {
        const int row = mtile * 16 + rbase + rr;
        qp[(size_t)row * U + col] = acc[rr] + bias;
    }
}

// ---------------------------------------------------------------------------
// Kernel 2: fused  score[g,b,t] = sum_u V[g,u]*tanh(qp[g,b,u] + b2[g,u]
//                                   + values[g,b,t,:].W2[g,:,u]) + bV[g]
// One workgroup = one (g,b) pair x 64-row T-tile. values tile staged in LDS
// via CDNA5 async global->LDS copies (padded stride). 8 waves; each wave owns
// 4 N-tiles and carries 4 M-tile accumulators through the K loop. The [T,U]
// intermediate never leaves registers.
// ---------------------------------------------------------------------------
__global__ __launch_bounds__(256) void score_fused_kernel(
    const float* __restrict__ values,  // [G*B, T, D]
    const float* __restrict__ W2,      // [G, D, U]
    const float* __restrict__ b2,      // [G, U]
    const float* __restrict__ Vv,      // [G, U]
    const float* __restrict__ bV,      // [G]
    const float* __restrict__ qp,      // [G*B, U]
    float* __restrict__ score)         // [G*B, T]
{
    extern __shared__ float smem[];
    float* sA     = smem;               // TT * LDA floats (132 KB)
    float* sQP    = sA + TT * LDA;      // U floats (qp + b2 combined)
    float* sV     = sQP + U;            // U floats
    float* sScore = sV + U;             // TT floats

    const int gb    = blockIdx.x;       // 0..95  (g*B + b)
    const int ttile = blockIdx.y;       // 0..15
    const int g     = gb >> 5;          // B = 32
    const int t0    = ttile * TT;
    const int tid   = threadIdx.x;

    const float* vals = values + ((size_t)gb * T + t0) * D;

    // Stage the 64x512 values tile into LDS.
#ifdef USE_ASYNC_STAGE
    for (int idx = tid; idx < TT * (D / 4); idx += 256) {
        const int row = idx >> 7;       // D/4 = 128 float4 per row
        const int c4  = idx & 127;
        __builtin_amdgcn_global_load_async_to_lds_b128(
            (as1_v4i_ptr)(void*)(vals + (size_t)row * D + c4 * 4),
            (as3_v4i_ptr)(void*)&sA[row * LDA + c4 * 4],
            0, 0);
    }
#else
    for (int idx = tid; idx < TT * (D / 4); idx += 256) {
        const int row = idx >> 7;
        const int c4  = idx & 127;
        float4 v = ((const float4*)(vals + (size_t)row * D))[c4];
        *(float4*)&sA[row * LDA + c4 * 4] = v;
    }
#endif
    for (int i = tid; i < U; i += 256) {
        sQP[i] = qp[(size_t)gb * U + i] + b2[(size_t)g * U + i];
        sV[i]  = Vv[(size_t)g * U + i];
    }
    if (tid < TT) sScore[tid] = 0.0f;
#ifdef USE_ASYNC_STAGE
    wait_async_zero();                  // own wave's async copies complete
#endif
    __syncthreads();                    // ...and all other waves' too

    const int wave = tid >> 5;
    const int lane = tid & 31;
    const int r    = lane & 15;
    const int kh   = lane >> 4;
    const float* Bg = W2 + (size_t)g * D * U;

    for (int j = 0; j < 4; ++j) {
        const int nt  = wave * 4 + j;       // this wave's N-tile
        const int col = nt * 16 + r;

        v8f acc[4] = {};
        #pragma unroll 2
        for (int k = 0; k < D; k += 4) {
            const int ka = k + 2 * kh;
            v2f b; b.x = Bg[(size_t)ka * U + col];
                   b.y = Bg[(size_t)(ka + 1) * U + col];
            #pragma unroll
            for (int mt = 0; mt < 4; ++mt) {
                v2f a = *(const v2f*)&sA[(mt * 16 + r) * LDA + ka];
                acc[mt] = __builtin_amdgcn_wmma_f32_16x16x4_f32(
                    false, a, false, b, (short)0, acc[mt], false, false);
            }
        }

        // Epilogue: tanh(+qp) * V, reduce over the 16 u-columns of this tile.
        const float qpu = sQP[col];
        const float vu  = sV[col];
        #pragma unroll
        for (int mt = 0; mt < 4; ++mt) {
            #pragma unroll
            for (int rr = 0; rr < 8; ++rr) {
                float f = tanhf(acc[mt][rr] + qpu) * vu;
                // reduce across each 16-lane half (same t, different u)
                #pragma unroll
                for (int off = 1; off < 16; off <<= 1)
                    f += __shfl_xor(f, off, 32);
                if (r == 0) {
                    const int tl = mt * 16 + kh * 8 + rr;
                    atomicAdd(&sScore[tl], f);     // ds_add_f32
                }
            }
        }
    }
    __syncthreads();

    if (tid < TT)
        score[(size_t)gb * T + t0 + tid] = sScore[tid] + bV[g];
}

// ---------------------------------------------------------------------------
// Kernel 3: per-(g,b) softmax over T, emit weights, then context = w^T.values
// (bandwidth-bound pass: coalesced float2 streams + global prefetch hints)
// ---------------------------------------------------------------------------
__global__ __launch_bounds__(256) void softmax_context_kernel(
    const float* __restrict__ values,  // [G*B, T, D]
    const float* __restrict__ score,   // [G*B, T]
    float* __restrict__ ctx,           // [G*B, D]
    float* __restrict__ wout)          // [G*B, T]
{
    __shared__ float sw[T];
    __shared__ float sred[256];

    const int gb  = blockIdx.x;
    const int tid = threadIdx.x;
    const float* sc = score + (size_t)gb * T;

    float lmax = -INFINITY;
    #pragma unroll
    for (int i = 0; i < T / 256; ++i) {
        const float s = sc[tid + i * 256];
        sw[tid + i * 256] = s;
        lmax = fmaxf(lmax, s);
    }
    sred[tid] = lmax;
    __syncthreads();
    for (int s = 128; s > 0; s >>= 1) {
        if (tid < s) sred[tid] = fmaxf(sred[tid], sred[tid + s]);
        __syncthreads();
    }
    const float m = sred[0];
    __syncthreads();

    float lsum = 0.0f;
    #pragma unroll
    for (int i = 0; i < T / 256; ++i) {
        const float e = __expf(sw[tid + i * 256] - m);
        sw[tid + i * 256] = e;
        lsum += e;
    }
    sred[tid] = lsum;
    __syncthreads();
    for (int s = 128; s > 0; s >>= 1) {
        if (tid < s) sred[tid] += sred[tid + s];
        __syncthreads();
    }
    const float inv = 1.0f / sred[0];
    __syncthreads();

    float* wo = wout + (size_t)gb * T;
    #pragma unroll
    for (int i = 0; i < T / 256; ++i) {
        const int t = tid + i * 256;
        const float w = sw[t] * inv;
        sw[t] = w;
        wo[t] = w;
    }
    __syncthreads();

    // context: each thread owns 2 adjacent feature columns (coalesced float2)
    const float* vals = values + (size_t)gb * T * D;
    const int d0 = tid * 2;
    float acc0 = 0.0f, acc1 = 0.0f;
    for (int t = 0; t < T; ++t) {
        if (t + 16 < T)
            __builtin_prefetch(vals + (size_t)(t + 16) * D + d0, 0, 1);
        const float2 v = *(const float2*)(vals + (size_t)t * D + d0);
        const float w = sw[t];
        acc0 = fmaf(w, v.x, acc0);
        acc1 = fmaf(w, v.y, acc1);
    }
    ctx[(size_t)gb * D + d0]     = acc0;
    ctx[(size_t)gb * D + d0 + 1] = acc1;
}

// ---------------------------------------------------------------------------
extern "C" void kernel_launch(void* const* d_in, const int* in_sizes, int n_in,
                              void* d_out, int out_size, void* d_ws, size_t ws_size,
                              hipStream_t stream) {
    (void)in_sizes; (void)n_in; (void)out_size; (void)ws_size;

    const float* query  = (const float*)d_in[0];   // [G,B,D]
    const float* values = (const float*)d_in[1];   // [G,B,T,D]
    const float* W1     = (const float*)d_in[2];   // [G,D,U]
    const float* b1     = (const float*)d_in[3];   // [G,U]
    const float* W2     = (const float*)d_in[4];   // [G,D,U]
    const float* b2     = (const float*)d_in[5];   // [G,U]
    const float* Vv     = (const float*)d_in[6];   // [G,U]
    const float* bV     = (const float*)d_in[7];   // [G]

    float* out     = (float*)d_out;
    float* ctx_out = out;                          // [G,B,D]
    float* w_out   = out + (size_t)G * Bc * D;     // [G,B,T,1]

    float* qp    = (float*)d_ws;                   // [G*B, U]
    float* score = qp + (size_t)G * Bc * U;        // [G*B, T]

    // K1: qp = query.W1 + b1   (6 x 32 tiles, 1 wave each)
    qp_wmma_kernel<<<dim3(G * Bc / 16, U / 16), 32, 0, stream>>>(query, W1, b1, qp);

    // K2: fused projection+tanh+V-dot -> scores
    const size_t shmem = (size_t)(TT * LDA + U + U + TT) * sizeof(float);
    score_fused_kernel<<<dim3(G * Bc, T / TT), 256, shmem, stream>>>(
        values, W2, b2, Vv, bV, qp, score);

    // K3: softmax over T + context
    softmax_context_kernel<<<G * Bc, 256, 0, stream>>>(values, score, ctx_out, w_out);
}